// LinearRegressionModel_240518168593
// MI455X (gfx1250) — compile-verified
//
#include <hip/hip_runtime.h>
#include <hip/hip_bf16.h>
#include <stdint.h>

// out[b, d] = bias[d] + sum_{t=0..7} W[d, t*1024 + x[b,t]]
// B=1024, CTX=8, VOCAB=1024, D=8192. Pure bandwidth problem (0.23 FLOP/byte):
// stream W exactly once via CDNA5 async global->LDS DMA, double-buffered per
// context band, gather columns from LDS.

#define VOCABSZ 1024
#define CTXSZ   8
#define DIMSZ   8192
#define BATCHSZ 1024
#define DT      32            // d-rows per block tile
#define PADROW  1028          // 1024 + 4 floats padding (bank rotation, 16B align)

__global__ __launch_bounds__(1024, 1)
void onehot_linear_gather_kernel(const int* __restrict__ x,
                                 const float* __restrict__ W,
                                 const float* __restrict__ bias,
                                 float* __restrict__ out) {
    // Double-buffered staging of W[d0:d0+DT, t*1024:(t+1)*1024]
    __shared__ float tile[2][DT * PADROW];   // 2 * 32 * 1028 * 4 B = 263168 B

    const unsigned tid = threadIdx.x;        // thread == batch index b
    const unsigned d0  = blockIdx.x * DT;

    // LDS byte offsets of the two buffers (flat ptr low 32 bits == LDS offset)
    const unsigned lds0 = (unsigned)(uintptr_t)(&tile[0][0]);
    const unsigned lds1 = (unsigned)(uintptr_t)(&tile[1][0]);

    // This thread's 8 context indices (two 16B loads).
    const int4* xp = (const int4*)(x + tid * CTXSZ);
    int4 xi0 = xp[0];
    int4 xi1 = xp[1];
    int idx[CTXSZ] = { xi0.x, xi0.y, xi0.z, xi0.w, xi1.x, xi1.y, xi1.z, xi1.w };

    // Issue the async DMA for context band t into buffer at lds_base.
    // Tile = DT rows x 1024 cols f32 = 8192 x 16B chunks, 8 chunks per thread.
    // chunk = tid + k*1024 -> consecutive lanes fetch consecutive 16B: fully
    // coalesced global side; LDS side lands row-major with PADROW stride.
    auto issue_band = [&](unsigned t, unsigned lds_base) {
        #pragma unroll
        for (unsigned k = 0; k < 8; ++k) {
            unsigned chunk = tid + k * 1024u;
            unsigned row   = chunk >> 8;                  // 256 chunks per row
            unsigned col4  = (chunk & 255u) << 2;         // float index in row
            unsigned ldso  = lds_base + (row * PADROW + col4) * 4u;
            unsigned goff  = ((d0 + row) * (unsigned)DIMSZ + t * VOCABSZ + col4) * 4u;
            asm volatile("global_load_async_to_lds_b128 %0, %1, %2"
                         :: "v"(ldso), "v"(goff), "s"(W)
                         : "memory");
        }
    };

    float acc[DT];
    #pragma unroll
    for (int j = 0; j < DT; ++j) acc[j] = 0.0f;

    // Prime the pipeline with band t=0.
    issue_band(0u, lds0);

    #pragma unroll
    for (int t = 0; t < CTXSZ; ++t) {
        if (t + 1 < CTXSZ) {
            // Prefetch next band into the other buffer, then wait only for the
            // first 8 (current band) async ops: ASYNCcnt <= 8 leaves the
            // prefetch in flight (async loads complete in order).
            issue_band((unsigned)(t + 1), (t & 1) ? lds0 : lds1);
            asm volatile("s_wait_asynccnt 0x8" ::: "memory");
        } else {
            asm volatile("s_wait_asynccnt 0x0" ::: "memory");
        }
        __syncthreads();   // all waves' portions of band t are in LDS

        // Gather column c = x[b,t] of the staged band: LDS random access.
        const int c = idx[t];
        const float* src = &tile[t & 1][c];
        #pragma unroll
        for (int j = 0; j < DT; ++j) {
            acc[j] += src[j * PADROW];
        }

        __syncthreads();   // everyone done reading before buffer is re-filled
    }

    // out[b, d0:d0+DT] : 128 contiguous bytes per thread (full cachelines).
    float* op = out + (size_t)tid * DIMSZ + d0;
    #pragma unroll
    for (int j4 = 0; j4 < DT / 4; ++j4) {
        float4 v;
        v.x = acc[4 * j4 + 0] + bias[d0 + 4 * j4 + 0];
        v.y = acc[4 * j4 + 1] + bias[d0 + 4 * j4 + 1];
        v.z = acc[4 * j4 + 2] + bias[d0 + 4 * j4 + 2];
        v.w = acc[4 * j4 + 3] + bias[d0 + 4 * j4 + 3];
        *(float4*)(op + 4 * j4) = v;
    }
}

extern "C" void kernel_launch(void* const* d_in, const int* in_sizes, int n_in,
                              void* d_out, int out_size, void* d_ws, size_t ws_size,
                              hipStream_t stream) {
    (void)in_sizes; (void)n_in; (void)out_size; (void)d_ws; (void)ws_size;
    const int*   x    = (const int*)d_in[0];    // [1024, 8] int32
    const float* W    = (const float*)d_in[1];  // [8192, 8192] f32 row-major
    const float* bias = (const float*)d_in[2];  // [8192] f32
    float*       out  = (float*)d_out;          // [1024, 8192] f32

    dim3 grid(DIMSZ / DT);   // 256 blocks, one 32-row d-tile each
    dim3 block(BATCHSZ);     // 1024 threads = 32 wave32, thread == batch
    hipLaunchKernelGGL(onehot_linear_gather_kernel, grid, block, 0, stream,
                       x, W, bias, out);
}